// GatedGCNNet_40699110097236
// MI455X (gfx1250) — compile-verified
//
#include <hip/hip_runtime.h>
#include <math.h>

// ---------------------------------------------------------------------------
// GatedGCN for MI455X (gfx1250).
//  - GEMMs (h @ W + b) use V_WMMA_F32_16X16X4_F32 (true fp32 matrix op) so
//    numerics match the fp32 reference exactly (modulo summation order).
//  - Edge message passing (the memory-bound phase, ~1.6 GB of traffic) uses
//    one wave32 per edge, float4 row loads, and L2 float atomics for the
//    segment sums (num/den fit easily in the 192MB L2).
// ---------------------------------------------------------------------------

typedef __attribute__((ext_vector_type(2))) float v2f;
typedef __attribute__((ext_vector_type(8))) float v8f;

#define NNODES  10000
#define NEDGES  640000
#define DFEAT   128
#define NLAYERS 4
#define NGRAPH  128
#define NCLASS  10

// ---------------------------------------------------------------------------
// Y[nrows, ncols] = X[nrows, 128] @ W[128, ncols] + bias[ncols]
// One wave computes a 16x64 tile: 4 v8f accumulators, K stepped by 4 using
// V_WMMA_F32_16X16X4_F32. fp32 WMMA fragment layouts (wave32):
//   A 16x4 : lanes 0-15 -> M=lane, v0=K0 v1=K1 ; lanes 16-31 -> v0=K2 v1=K3
//   B 4x16 : lanes 0-15 -> N=lane, v0=K0 v1=K1 ; lanes 16-31 -> v0=K2 v1=K3
//   C 16x16: vr -> M=r (lanes 0-15) / M=r+8 (lanes 16-31), N=lane%16
// nrows must be a multiple of 16 (10000 = 625*16). Uniform wave-level guard
// only, so EXEC stays all-ones for WMMA.
// ---------------------------------------------------------------------------
__global__ __launch_bounds__(256)
void gemm_bias_wmma(const float* __restrict__ X, const float* __restrict__ W,
                    const float* __restrict__ bias, float* __restrict__ Y,
                    int nrows, int ncols) {
    const int wave = threadIdx.x >> 5;
    const int lane = threadIdx.x & 31;
    const int lm   = lane & 15;
    const int lh   = lane >> 4;

    const int rowTile = blockIdx.x * 8 + wave;   // wave-uniform
    const int row0    = rowTile * 16;
    if (row0 >= nrows) return;                   // uniform exit, EXEC stays full
    const int col0 = blockIdx.y * 64;

    v8f acc[4] = {};
    const float* xrow = X + (size_t)(row0 + lm) * DFEAT;

    for (int k = 0; k < DFEAT; k += 4) {
        const int ka = k + 2 * lh;
        const v2f a = *(const v2f*)(xrow + ka);            // A fragment (b64 load)
        const float* wp0 = W + (size_t)ka * ncols + col0 + lm;
        const float* wp1 = wp0 + ncols;
        v2f b[4];
        #pragma unroll
        for (int j = 0; j < 4; ++j) {
            b[j].x = wp0[16 * j];
            b[j].y = wp1[16 * j];
        }
        #pragma unroll
        for (int j = 0; j < 4; ++j) {
            acc[j] = __builtin_amdgcn_wmma_f32_16x16x4_f32(
                false, a, false, b[j], (short)0, acc[j], false, false);
        }
    }

    #pragma unroll
    for (int j = 0; j < 4; ++j) {
        const int col = col0 + 16 * j + lm;
        const float bb = bias[col];
        #pragma unroll
        for (int r = 0; r < 8; ++r) {
            Y[(size_t)(row0 + r + 8 * lh) * ncols + col] = acc[j][r] + bb;
        }
    }
}

// ---------------------------------------------------------------------------
// One wave per edge: sigma = sigmoid(Dh[dst] + Eh[src]);
// num[dst] += sigma * Bh[src]; den[dst] += sigma. 4 features per lane.
// ---------------------------------------------------------------------------
__device__ __forceinline__ float sigmoidf_(float x) {
    return 1.0f / (1.0f + expf(-x));
}

__global__ __launch_bounds__(256)
void edge_gate_kernel(const int* __restrict__ ei, const float* __restrict__ Bh,
                      const float* __restrict__ Dh, const float* __restrict__ Eh,
                      float* __restrict__ num, float* __restrict__ den) {
    const int t    = blockIdx.x * blockDim.x + threadIdx.x;
    const int e    = t >> 5;
    const int lane = t & 31;
    if (e >= NEDGES) return;

    const int src = ei[e];
    const int dst = ei[NEDGES + e];
    const int c0  = lane * 4;

    const float4 d4 = *(const float4*)(Dh + (size_t)dst * DFEAT + c0);
    const float4 e4 = *(const float4*)(Eh + (size_t)src * DFEAT + c0);
    const float4 b4 = *(const float4*)(Bh + (size_t)src * DFEAT + c0);

    const float s0 = sigmoidf_(d4.x + e4.x);
    const float s1 = sigmoidf_(d4.y + e4.y);
    const float s2 = sigmoidf_(d4.z + e4.z);
    const float s3 = sigmoidf_(d4.w + e4.w);

    float* np = num + (size_t)dst * DFEAT + c0;
    float* dp = den + (size_t)dst * DFEAT + c0;
    atomicAdd(np + 0, s0 * b4.x);
    atomicAdd(np + 1, s1 * b4.y);
    atomicAdd(np + 2, s2 * b4.z);
    atomicAdd(np + 3, s3 * b4.w);
    atomicAdd(dp + 0, s0);
    atomicAdd(dp + 1, s1);
    atomicAdd(dp + 2, s2);
    atomicAdd(dp + 3, s3);
}

// ---------------------------------------------------------------------------
// h_new = Ah + num/(den+1e-6) -> tmp, plus per-column sum / sumsq for BN.
// Block covers 128 rows x 128 cols; 2 threads per column, LDS reduce,
// then one atomicAdd per column per block.
// ---------------------------------------------------------------------------
__global__ __launch_bounds__(256)
void bn_stats_kernel(const float* __restrict__ Ah, const float* __restrict__ num,
                     const float* __restrict__ den, float* __restrict__ tmp,
                     float* __restrict__ sums, float* __restrict__ sumsq) {
    __shared__ float ls[256];
    __shared__ float lq[256];
    const int col  = threadIdx.x & 127;
    const int rsub = threadIdx.x >> 7;
    const int base = blockIdx.x * 128;

    float s = 0.0f, q = 0.0f;
    for (int r = rsub; r < 128; r += 2) {
        const int i = base + r;
        if (i < NNODES) {
            const size_t off = (size_t)i * DFEAT + col;
            const float v = Ah[off] + num[off] / (den[off] + 1e-6f);
            tmp[off] = v;
            s += v;
            q += v * v;
        }
    }
    ls[threadIdx.x] = s;
    lq[threadIdx.x] = q;
    __syncthreads();
    if (rsub == 0) {
        atomicAdd(&sums[col],  ls[col] + ls[col + 128]);
        atomicAdd(&sumsq[col], lq[col] + lq[col + 128]);
    }
}

// h += relu(gamma*(tmp-mean)*rsqrt(var+1e-5)+beta)   (biased var, as jnp.var)
__global__ __launch_bounds__(256)
void bn_apply_kernel(const float* __restrict__ tmp, const float* __restrict__ sums,
                     const float* __restrict__ sumsq, const float* __restrict__ gamma,
                     const float* __restrict__ beta, float* __restrict__ h) {
    const size_t i = (size_t)blockIdx.x * blockDim.x + threadIdx.x;
    if (i >= (size_t)NNODES * DFEAT) return;
    const int col = (int)(i & (DFEAT - 1));
    const float invN = 1.0f / (float)NNODES;
    const float mean = sums[col] * invN;
    const float var  = sumsq[col] * invN - mean * mean;
    const float v = gamma[col] * (tmp[i] - mean) * rsqrtf(var + 1e-5f) + beta[col];
    h[i] += fmaxf(v, 0.0f);
}

// Per-graph mean readout accumulation: one wave per node.
__global__ __launch_bounds__(256)
void readout_accum(const float* __restrict__ h, const int* __restrict__ batch,
                   float* __restrict__ hg, float* __restrict__ counts) {
    const int t    = blockIdx.x * blockDim.x + threadIdx.x;
    const int i    = t >> 5;
    const int lane = t & 31;
    if (i >= NNODES) return;
    const int g  = batch[i];
    const int c0 = lane * 4;
    const float4 v = *(const float4*)(h + (size_t)i * DFEAT + c0);
    float* p = hg + (size_t)g * DFEAT + c0;
    atomicAdd(p + 0, v.x);
    atomicAdd(p + 1, v.y);
    atomicAdd(p + 2, v.z);
    atomicAdd(p + 3, v.w);
    if (lane == 0) atomicAdd(counts + g, 1.0f);
}

// MLP readout: 128 -> 64 -> 32 -> 10, one block per graph (tiny).
__global__ __launch_bounds__(64)
void mlp_readout(const float* __restrict__ hg, const float* __restrict__ counts,
                 const float* __restrict__ w0, const float* __restrict__ b0,
                 const float* __restrict__ w1, const float* __restrict__ b1,
                 const float* __restrict__ w2, const float* __restrict__ b2,
                 float* __restrict__ out) {
    __shared__ float xin[128];
    __shared__ float x0[64];
    __shared__ float x1[32];
    const int g = blockIdx.x;
    const int t = threadIdx.x;

    const float inv = 1.0f / fmaxf(counts[g], 1.0f);
    xin[t]      = hg[(size_t)g * DFEAT + t] * inv;
    xin[t + 64] = hg[(size_t)g * DFEAT + t + 64] * inv;
    __syncthreads();

    {
        float s = b0[t];
        for (int k = 0; k < 128; ++k) s += xin[k] * w0[k * 64 + t];
        x0[t] = fmaxf(s, 0.0f);
    }
    __syncthreads();
    if (t < 32) {
        float s = b1[t];
        for (int k = 0; k < 64; ++k) s += x0[k] * w1[k * 32 + t];
        x1[t] = fmaxf(s, 0.0f);
    }
    __syncthreads();
    if (t < NCLASS) {
        float s = b2[t];
        for (int k = 0; k < 32; ++k) s += x1[k] * w2[k * NCLASS + t];
        out[(size_t)g * NCLASS + t] = s;
    }
}

// ---------------------------------------------------------------------------
extern "C" void kernel_launch(void* const* d_in, const int* in_sizes, int n_in,
                              void* d_out, int out_size, void* d_ws, size_t ws_size,
                              hipStream_t stream) {
    (void)in_sizes; (void)n_in; (void)out_size; (void)ws_size;

    const float* feature = (const float*)d_in[0];
    const int*   edge_ix = (const int*)  d_in[1];
    const int*   batch   = (const int*)  d_in[2];
    const float* embed_w = (const float*)d_in[3];
    const float* embed_b = (const float*)d_in[4];
    const float* Aw = (const float*)d_in[5];
    const float* Ab = (const float*)d_in[6];
    const float* Bw = (const float*)d_in[7];
    const float* Bb = (const float*)d_in[8];
    const float* Dw = (const float*)d_in[9];
    const float* Db = (const float*)d_in[10];
    const float* Ew = (const float*)d_in[11];
    const float* Eb = (const float*)d_in[12];
    const float* bn_g = (const float*)d_in[13];
    const float* bn_b = (const float*)d_in[14];
    const float* mlp_w0 = (const float*)d_in[15];
    const float* mlp_b0 = (const float*)d_in[16];
    const float* mlp_w1 = (const float*)d_in[17];
    const float* mlp_b1 = (const float*)d_in[18];
    const float* mlp_w2 = (const float*)d_in[19];
    const float* mlp_b2 = (const float*)d_in[20];
    float* out = (float*)d_out;

    // Workspace layout (floats): ~41 MB total.
    const size_t ND = (size_t)NNODES * DFEAT;
    float* ws    = (float*)d_ws;
    float* h     = ws;                 // [N,D]
    float* Ah    = h   + ND;           // [N,D]
    float* Bh    = Ah  + ND;           // [N,D]
    float* Dh    = Bh  + ND;           // [N,D]
    float* Eh    = Dh  + ND;           // [N,D]
    float* numb  = Eh  + ND;           // [N,D]  (num; adjacent to den for one memset)
    float* den   = numb + ND;          // [N,D]
    float* tmp   = den + ND;           // [N,D]  h_new pre-BN
    float* sums  = tmp + ND;           // [D]
    float* sumsq = sums + DFEAT;       // [D]
    float* hg    = sumsq + DFEAT;      // [G,D]  (adjacent to counts for one memset)
    float* cnts  = hg + (size_t)NGRAPH * DFEAT;  // [G]

    const int rowTiles = NNODES / 16;                 // 625 (exact)
    dim3 gemmGrid((rowTiles + 7) / 8, DFEAT / 64);    // (79, 2)

    const int edgeBlocks  = (NEDGES * 32 + 255) / 256;     // 80000
    const int statsBlocks = (NNODES + 127) / 128;          // 79
    const int applyBlocks = (int)((ND + 255) / 256);
    const int rdBlocks    = (NNODES * 32 + 255) / 256;

    // h = feature @ embed_w + embed_b
    gemm_bias_wmma<<<gemmGrid, 256, 0, stream>>>(feature, embed_w, embed_b, h,
                                                 NNODES, DFEAT);

    for (int l = 0; l < NLAYERS; ++l) {
        const size_t wo = (size_t)l * DFEAT * DFEAT;
        const size_t bo = (size_t)l * DFEAT;
        gemm_bias_wmma<<<gemmGrid, 256, 0, stream>>>(h, Aw + wo, Ab + bo, Ah, NNODES, DFEAT);
        gemm_bias_wmma<<<gemmGrid, 256, 0, stream>>>(h, Bw + wo, Bb + bo, Bh, NNODES, DFEAT);
        gemm_bias_wmma<<<gemmGrid, 256, 0, stream>>>(h, Dw + wo, Db + bo, Dh, NNODES, DFEAT);
        gemm_bias_wmma<<<gemmGrid, 256, 0, stream>>>(h, Ew + wo, Eb + bo, Eh, NNODES, DFEAT);

        hipMemsetAsync(numb, 0, 2 * ND * sizeof(float), stream);      // num + den
        hipMemsetAsync(sums, 0, 2 * DFEAT * sizeof(float), stream);   // sums + sumsq

        edge_gate_kernel<<<edgeBlocks, 256, 0, stream>>>(edge_ix, Bh, Dh, Eh, numb, den);
        bn_stats_kernel<<<statsBlocks, 256, 0, stream>>>(Ah, numb, den, tmp, sums, sumsq);
        bn_apply_kernel<<<applyBlocks, 256, 0, stream>>>(tmp, sums, sumsq,
                                                         bn_g + bo, bn_b + bo, h);
    }

    hipMemsetAsync(hg, 0, ((size_t)NGRAPH * DFEAT + NGRAPH) * sizeof(float), stream);
    readout_accum<<<rdBlocks, 256, 0, stream>>>(h, batch, hg, cnts);
    mlp_readout<<<NGRAPH, 64, 0, stream>>>(hg, cnts, mlp_w0, mlp_b0,
                                           mlp_w1, mlp_b1, mlp_w2, mlp_b2, out);
}